// GraphAttentionLayer_55233279426987
// MI455X (gfx1250) — compile-verified
//
#include <hip/hip_runtime.h>
#include <math.h>

#define DIM 128

typedef __attribute__((ext_vector_type(2))) float v2f;
typedef __attribute__((ext_vector_type(8))) float v8f;

// ---- order-preserving float <-> uint mapping for atomic max on f32 ----
__device__ __forceinline__ unsigned orderedF(float f) {
  unsigned u = __float_as_uint(f);
  return (u & 0x80000000u) ? ~u : (u | 0x80000000u);
}
__device__ __forceinline__ float unorderedF(unsigned u) {
  return (u & 0x80000000u) ? __uint_as_float(u & 0x7FFFFFFFu)
                           : __uint_as_float(~u);
}

// -------------------------------------------------------------------------
// out[r] = dot(X[r,:], w[0:128]) for n_rows rows, via V_WMMA_F32_16X16X4_F32.
// One wave handles 16 rows; B tile = weight chunk replicated across all 16
// columns, so every column of D holds the same dot product.
// Layouts (ISA 7.12.2, wave32, h = lane>>4, v = vgpr index):
//   A 16x4 f32 : lane m=lane&15 holds row m, K = 2h+v
//   B 4x16 f32 : K = v+2h (N = lane&15, value independent of N here)
//   D 16x16    : M = v+8h, N = lane&15
// -------------------------------------------------------------------------
__global__ void dot_rows_wmma(const float* __restrict__ X,
                              const float* __restrict__ w,
                              float* __restrict__ out, int n_rows) {
  int warp = (blockIdx.x * blockDim.x + threadIdx.x) >> 5;
  int lane = threadIdx.x & 31;
  int row_base = warp * 16;
  if (row_base >= n_rows) return;  // warp-uniform: EXEC all-1s inside

  int m = lane & 15;
  int h = lane >> 4;
  const float* xr = X + (size_t)(row_base + m) * DIM;

  v8f c = {0.f, 0.f, 0.f, 0.f, 0.f, 0.f, 0.f, 0.f};
#if __has_builtin(__builtin_amdgcn_wmma_f32_16x16x4_f32)
#pragma unroll
  for (int k0 = 0; k0 < DIM; k0 += 4) {
    v2f a, b;
    a.x = xr[k0 + 2 * h];
    a.y = xr[k0 + 2 * h + 1];
    b.x = w[k0 + 2 * h];
    b.y = w[k0 + 2 * h + 1];
    c = __builtin_amdgcn_wmma_f32_16x16x4_f32(false, a, false, b,
                                              (short)0, c, false, false);
  }
  if (m == 0) {
#pragma unroll
    for (int v = 0; v < 8; ++v) out[row_base + 8 * h + v] = c[v];
  }
#else
  // VALU fallback (should not be taken on gfx1250)
  float acc = 0.f;
  for (int k = lane; k < DIM; k += 32) acc += xr[k] * 0.f;  // placeholder
  (void)acc; (void)c;
  if (lane < 16) {
    float s = 0.f;
    const float* r = X + (size_t)(row_base + lane) * DIM;
    for (int k = 0; k < DIM; ++k) s += r[k] * w[k];
    out[row_base + lane] = s;
  }
#endif
}

// -------------------------------------------------------------------------
__global__ void init_kernel(float* __restrict__ out, float* __restrict__ denom,
                            unsigned* __restrict__ m_u, int n_nodes) {
  int i = blockIdx.x * blockDim.x + threadIdx.x;
  int total = n_nodes * DIM;
  if (i < total) out[i] = 0.f;
  if (i < n_nodes) {
    denom[i] = 0.f;
    m_u[i] = orderedF(-INFINITY);
  }
}

// per-incidence score -> atomic segment max
__global__ void scatter_max(const int* __restrict__ el,
                            const float* __restrict__ s_node,
                            const float* __restrict__ s_edge,
                            const float* __restrict__ bp,
                            unsigned* __restrict__ m_u, int n_edges) {
  int i = blockIdx.x * blockDim.x + threadIdx.x;
  if (i >= 2 * n_edges) return;
  int e = (i < n_edges) ? i : i - n_edges;
  int seg = (i < n_edges) ? el[2 * e] : el[2 * e + 1];
  float raw = s_node[seg] + s_edge[e] + bp[0];
  float score = raw >= 0.f ? raw : 0.2f * raw;
  atomicMax(m_u + seg, orderedF(score));
}

// one wave per incidence: ex = exp(score - m), denom += ex, out += ex * ee
__global__ void scatter_add(const int* __restrict__ el,
                            const float* __restrict__ s_node,
                            const float* __restrict__ s_edge,
                            const float* __restrict__ bp,
                            const unsigned* __restrict__ m_u,
                            const float4* __restrict__ ee4,
                            float* __restrict__ out,
                            float* __restrict__ denom, int n_edges) {
  int gid = blockIdx.x * blockDim.x + threadIdx.x;
  int i = gid >> 5;
  int lane = gid & 31;
  if (i >= 2 * n_edges) return;
  int e = (i < n_edges) ? i : i - n_edges;
  int seg = (i < n_edges) ? el[2 * e] : el[2 * e + 1];
  float raw = s_node[seg] + s_edge[e] + bp[0];
  float score = raw >= 0.f ? raw : 0.2f * raw;
  float ex = expf(score - unorderedF(m_u[seg]));
  if (lane == 0) atomicAdd(denom + seg, ex);
  float4 v = ee4[(size_t)e * (DIM / 4) + lane];
  float* o = out + (size_t)seg * DIM + lane * 4;
  atomicAdd(o + 0, ex * v.x);
  atomicAdd(o + 1, ex * v.y);
  atomicAdd(o + 2, ex * v.z);
  atomicAdd(o + 3, ex * v.w);
}

// normalize + exact GELU, or pass through node_emb for degree-0 nodes
__global__ void finalize(const float* __restrict__ node_emb,
                         const float* __restrict__ denom,
                         float* __restrict__ out, int n_nodes) {
  int i = blockIdx.x * blockDim.x + threadIdx.x;
  if (i >= n_nodes * DIM) return;
  int n = i >> 7;  // / DIM
  float den = denom[n];
  if (den > 0.f) {
    float x = out[i] / den;
    out[i] = 0.5f * x * (1.f + erff(x * 0.70710678118654752440f));
  } else {
    out[i] = node_emb[i];
  }
}

// -------------------------------------------------------------------------
extern "C" void kernel_launch(void* const* d_in, const int* in_sizes, int n_in,
                              void* d_out, int out_size, void* d_ws,
                              size_t ws_size, hipStream_t stream) {
  const float* node_emb  = (const float*)d_in[0];
  const float* edge_emb  = (const float*)d_in[1];
  const float* W_att_w   = (const float*)d_in[2];
  const float* W_att_b   = (const float*)d_in[3];
  const int*   edge_list = (const int*)d_in[4];

  const int n_nodes = in_sizes[0] / DIM;   // 50000
  const int n_edges = in_sizes[1] / DIM;   // 800000

  float* out = (float*)d_out;
  float* ws = (float*)d_ws;
  float*    s_node = ws;                                   // n_nodes
  float*    s_edge = ws + n_nodes;                         // n_edges
  unsigned* m_u    = (unsigned*)(ws + n_nodes + n_edges);  // n_nodes
  float*    denom  = ws + 2 * (size_t)n_nodes + n_edges;   // n_nodes

  const int total_out = n_nodes * DIM;
  init_kernel<<<(total_out + 255) / 256, 256, 0, stream>>>(out, denom, m_u,
                                                           n_nodes);

  // s_node = node_emb @ w_n ; s_edge = edge_emb @ w_e  (WMMA f32 16x16x4)
  {
    int warps = (n_nodes + 15) / 16;
    int thr = warps * 32;
    dot_rows_wmma<<<(thr + 255) / 256, 256, 0, stream>>>(node_emb, W_att_w,
                                                         s_node, n_nodes);
  }
  {
    int warps = (n_edges + 15) / 16;
    int thr = warps * 32;
    dot_rows_wmma<<<(thr + 255) / 256, 256, 0, stream>>>(
        edge_emb, W_att_w + DIM, s_edge, n_edges);
  }

  scatter_max<<<(2 * n_edges + 255) / 256, 256, 0, stream>>>(
      edge_list, s_node, s_edge, W_att_b, m_u, n_edges);

  {
    long long thr = 2LL * n_edges * 32;
    scatter_add<<<(int)((thr + 255) / 256), 256, 0, stream>>>(
        edge_list, s_node, s_edge, W_att_b, m_u, (const float4*)edge_emb, out,
        denom, n_edges);
  }

  finalize<<<(total_out + 255) / 256, 256, 0, stream>>>(node_emb, denom, out,
                                                        n_nodes);
}